// latent_correlation_layer_87832081203633
// MI455X (gfx1250) — compile-verified
//
#include <hip/hip_runtime.h>

#define B_ 8
#define N_ 4096
#define F_ 128
#define D_ 4
#define NT_ (N_ / 16)   // 256 m-tiles per row-block

typedef float v2f __attribute__((ext_vector_type(2)));
typedef float v8f __attribute__((ext_vector_type(8)));

// scale by 1/sqrt(D)=0.5 then LeakyReLU(0.2); identity: leaky(s)=max(s,0.2s)
__device__ __forceinline__ float scale_leaky(float c) {
    const float s = c * 0.5f;
    return fmaxf(s, 0.2f * s);
}

// ---------------------------------------------------------------------------
// Kernel 1: q = x @ Wq, k = x @ Wk   (x:[B,N,F] f32, W:[F,4])
// One wave per row: lane loads float4 of x (coalesced), weights in LDS,
// butterfly shuffle reduction, lane 0 writes q[row][0..3], k[row][0..3].
// ---------------------------------------------------------------------------
__global__ void proj_qk_kernel(const float* __restrict__ x,
                               const float* __restrict__ wk,
                               const float* __restrict__ wq,
                               float* __restrict__ q,
                               float* __restrict__ k) {
    __shared__ float s_wq[F_ * D_];
    __shared__ float s_wk[F_ * D_];
    for (int i = threadIdx.x; i < F_ * D_; i += blockDim.x) {
        s_wq[i] = wq[i];
        s_wk[i] = wk[i];
    }
    __syncthreads();

    const int wave = threadIdx.x >> 5;          // 8 waves/block
    const int lane = threadIdx.x & 31;
    const int row  = blockIdx.x * 8 + wave;     // row in [0, B*N)
    if (row >= B_ * N_) return;

    const float4 xv = *(const float4*)(x + (size_t)row * F_ + lane * 4);
    float qa[D_] = {0.f, 0.f, 0.f, 0.f};
    float ka[D_] = {0.f, 0.f, 0.f, 0.f};
    const float xs[4] = {xv.x, xv.y, xv.z, xv.w};
#pragma unroll
    for (int j = 0; j < 4; ++j) {
        const int f = lane * 4 + j;
#pragma unroll
        for (int d = 0; d < D_; ++d) {
            qa[d] = fmaf(xs[j], s_wq[f * D_ + d], qa[d]);
            ka[d] = fmaf(xs[j], s_wk[f * D_ + d], ka[d]);
        }
    }
#pragma unroll
    for (int m = 16; m >= 1; m >>= 1) {
#pragma unroll
        for (int d = 0; d < D_; ++d) {
            qa[d] += __shfl_xor(qa[d], m, 32);
            ka[d] += __shfl_xor(ka[d], m, 32);
        }
    }
    if (lane == 0) {
#pragma unroll
        for (int d = 0; d < D_; ++d) {
            q[row * D_ + d] = qa[d];
            k[row * D_ + d] = ka[d];
        }
    }
}

// ---------------------------------------------------------------------------
// Kernel 2: fused scores -> LeakyReLU -> softmax -> batch-mean per 16-row
// n-tile. One block (8 waves) per tile row; WMMA f32 16x16x4 does each 16x16
// score tile (K = D = 4 in one instruction, full f32).
//
// Softmax uses LANE-LOCAL running stats (no in-loop shuffles; gfx1250 lowers
// __shfl_xor to ds_bpermute + dscnt waits). Three WMMA sweeps per batch:
//   sweep 1: per-lane row max            (1 v_max/elem)
//   sweep 2: per-lane sum of exp(s-max)  (1 v_exp/elem)
//   sweep 3 (b innermost): mean accumulated in registers, stored to global.
// Per-row stats for all 8 batches live in a 1KB LDS table; no big LDS
// accumulator, no LDS RMW per element, high occupancy.
//
// Fragment layouts (ISA 7.12.2, wave32):
//   A (16x4 f32): lanes 0-15 -> M=lane, VGPR0/1=K0/K1; lanes 16-31 -> K2/K3
//   B (4x16 f32): lanes 0-15 -> N=lane, VGPR0/1=K0/K1; lanes 16-31 -> K2/K3
//   C/D 16x16:    VGPR r: lanes 0-15 = (M=r, N=lane); lanes 16-31 = (M=8+r)
// ---------------------------------------------------------------------------
__global__ void attn_mean_kernel(const float* __restrict__ q,
                                 const float* __restrict__ k,
                                 float* __restrict__ A /* [N,N] */) {
    __shared__ float2 st[B_][16];    // .x = row max, .y = (1/rowsum) * (1/B)
    __shared__ float  red[8 * 16];   // per-wave partials for cross-wave combine

    const int wave = threadIdx.x >> 5;
    const int lane = threadIdx.x & 31;
    const int half = lane >> 4;      // 0: rows 0-7 / K0-1, 1: rows 8-15 / K2-3
    const int l15  = lane & 15;
    const int d0   = half * 2;
    const int n_base = blockIdx.x * 16;
    const float NEG_BIG = -3.0e38f;

    // Preload the A fragment (q rows of this n-tile) for all 8 batches.
    v2f afr[B_];
#pragma unroll
    for (int b = 0; b < B_; ++b)
        afr[b] = *(const v2f*)(q + (b * N_ + n_base + l15) * D_ + d0);

    // ================= stats phase: row max & 1/rowsum per batch ==========
    for (int b = 0; b < B_; ++b) {
        // ---- sweep 1: lane-local row max ----
        float runmax[8];
#pragma unroll
        for (int r = 0; r < 8; ++r) runmax[r] = NEG_BIG;

        for (int t = wave; t < NT_; t += 8) {
            const int col = b * N_ + t * 16 + l15;
            const v2f bfrag = *(const v2f*)(k + col * D_ + d0);
            // branchless prefetch of next k tile (clamped, CDNA5
            // global_prefetch_b8) -- no per-iteration EXEC branch
            const int pt = (t + 8 < NT_) ? (t + 8) : t;
            __builtin_prefetch(k + (b * N_ + pt * 16 + l15) * D_ + d0, 0, 1);
            v8f c = {};
            c = __builtin_amdgcn_wmma_f32_16x16x4_f32(
                    false, afr[b], false, bfrag, (short)0, c, false, false);
#pragma unroll
            for (int r = 0; r < 8; ++r)
                runmax[r] = fmaxf(runmax[r], scale_leaky(c[r]));
        }
        // single 16-lane-half butterfly per row, then cross-wave combine
        float gmax[8];
#pragma unroll
        for (int r = 0; r < 8; ++r) {
            float m = runmax[r];
#pragma unroll
            for (int msk = 1; msk <= 8; msk <<= 1)
                m = fmaxf(m, __shfl_xor(m, msk, 32));
            if (l15 == 0) red[wave * 16 + half * 8 + r] = m;
        }
        __syncthreads();
#pragma unroll
        for (int r = 0; r < 8; ++r) {
            float gm = NEG_BIG;
            for (int w = 0; w < 8; ++w)
                gm = fmaxf(gm, red[w * 16 + half * 8 + r]);
            gmax[r] = gm;
            if (wave == 0 && l15 == 0) st[b][half * 8 + r].x = gm;
        }
        __syncthreads();   // red[] reused below

        // ---- sweep 2: lane-local sum of exp(s - gmax) ----
        float rsum[8];
#pragma unroll
        for (int r = 0; r < 8; ++r) rsum[r] = 0.f;

        for (int t = wave; t < NT_; t += 8) {
            const int col = b * N_ + t * 16 + l15;
            const v2f bfrag = *(const v2f*)(k + col * D_ + d0);
            v8f c = {};
            c = __builtin_amdgcn_wmma_f32_16x16x4_f32(
                    false, afr[b], false, bfrag, (short)0, c, false, false);
#pragma unroll
            for (int r = 0; r < 8; ++r)
                rsum[r] += __expf(scale_leaky(c[r]) - gmax[r]);
        }
#pragma unroll
        for (int r = 0; r < 8; ++r) {
            float sm = rsum[r];
#pragma unroll
            for (int msk = 1; msk <= 8; msk <<= 1)
                sm += __shfl_xor(sm, msk, 32);
            if (l15 == 0) red[wave * 16 + half * 8 + r] = sm;
        }
        __syncthreads();
        if (wave == 0 && l15 == 0) {
#pragma unroll
            for (int r = 0; r < 8; ++r) {
                float gs = 0.f;
                for (int w = 0; w < 8; ++w) gs += red[w * 16 + half * 8 + r];
                st[b][half * 8 + r].y = 0.125f / gs;   // 1/sum * 1/B
            }
        }
        __syncthreads();
    }

    // ====== accumulate phase: batch-mean in registers, direct global store =
    for (int t = wave; t < NT_; t += 8) {
        float accm[8];
#pragma unroll
        for (int r = 0; r < 8; ++r) accm[r] = 0.f;

        for (int b = 0; b < B_; ++b) {
            const int col = b * N_ + t * 16 + l15;
            const v2f bfrag = *(const v2f*)(k + col * D_ + d0);
            v8f c = {};
            c = __builtin_amdgcn_wmma_f32_16x16x4_f32(
                    false, afr[b], false, bfrag, (short)0, c, false, false);
#pragma unroll
            for (int r = 0; r < 8; ++r) {
                const float2 sv = st[b][half * 8 + r];  // 64B/lane, contiguous
                accm[r] = fmaf(__expf(scale_leaky(c[r]) - sv.x), sv.y, accm[r]);
            }
        }
#pragma unroll
        for (int r = 0; r < 8; ++r)
            A[(size_t)(n_base + half * 8 + r) * N_ + t * 16 + l15] = accm[r];
    }
}

// ---------------------------------------------------------------------------
// Kernel 3: in-place symmetrize  out = 0.5*(A + A^T).  Each (i<j) pair owned
// by exactly one thread; diagonal is a fixed point (replay-deterministic).
// Whole matrix (67 MB) is L2-resident on MI455X (192 MB global L2), so the
// transposed reads stay on-chip.
// ---------------------------------------------------------------------------
__global__ void symmetrize_kernel(float* __restrict__ A) {
    const int idx = blockIdx.x * blockDim.x + threadIdx.x;
    if (idx >= N_ * N_) return;
    const int i = idx >> 12;           // / N_
    const int j = idx & (N_ - 1);
    if (i < j) {
        const float a = A[idx];
        const float b = A[(size_t)j * N_ + i];
        const float v = 0.5f * (a + b);
        A[idx] = v;
        A[(size_t)j * N_ + i] = v;
    }
}

// ---------------------------------------------------------------------------
extern "C" void kernel_launch(void* const* d_in, const int* in_sizes, int n_in,
                              void* d_out, int out_size, void* d_ws, size_t ws_size,
                              hipStream_t stream) {
    const float* x  = (const float*)d_in[0];   // [B,N,F]
    const float* wk = (const float*)d_in[1];   // [F,D]
    const float* wq = (const float*)d_in[2];   // [F,D]
    float* out = (float*)d_out;                // [N,N]

    float* q = (float*)d_ws;                   // [B*N, 4]
    float* k = q + (size_t)B_ * N_ * D_;       // [B*N, 4]  (ws usage: 1 MB)

    // K1: projections. wave-per-row, 8 rows per 256-thread block.
    proj_qk_kernel<<<(B_ * N_) / 8, 256, 0, stream>>>(x, wk, wq, q, k);

    // K2: fused WMMA attention + softmax + batch mean. ~1.5 KB static LDS.
    attn_mean_kernel<<<N_ / 16, 256, 0, stream>>>(q, k, out);

    // K3: in-place symmetrization.
    symmetrize_kernel<<<(N_ * N_ + 255) / 256, 256, 0, stream>>>(out);
}